// VectorQuantizer_57526791963060
// MI455X (gfx1250) — compile-verified
//
#include <hip/hip_runtime.h>
#include <hip/hip_bf16.h>

// ---------------- problem constants ----------------
#define BATCH   32
#define EMBED   128          // C / K-dim of GEMM
#define HW      1024         // 32*32
#define NROWS   32768        // BATCH*HW
#define NCODES  2048
#define TILE_M  32           // rows per workgroup (2 x 16-row WMMA subtiles)
#define SUB     16

// output offsets (flat float elements, reference tuple order)
#define OFF_LOSS    ((size_t)0)
#define OFF_ZQ      ((size_t)1)
#define OFF_PERP    ((size_t)4194305)
#define OFF_ONEHOT  ((size_t)4194306)
#define OFF_IDX     ((size_t)71303170)
#define OFF_IHIST   ((size_t)71335938)
#define OFF_SHIST   ((size_t)71401474)
#define OFF_COMMIT  ((size_t)71467010)
#define OFF_QUANT   ((size_t)71467011)
#define OFF_SLOSS   ((size_t)71467012)

#define STRIDE_ZT    130     // padded LDS stride for z tile (bank-conflict free b64 reads)
#define STRIDE_STRIP 2052    // padded LDS stride for dist strip

typedef __attribute__((ext_vector_type(2))) float v2f;
typedef __attribute__((ext_vector_type(8))) float v8f;

// -------- zero accumulators + both histograms ----------
__global__ __launch_bounds__(256) void vq_init(float* __restrict__ out,
                                               float* __restrict__ acc) {
    int i = blockIdx.x * 256 + threadIdx.x;
    if (i < 2) acc[i] = 0.0f;
    if (i < 2 * BATCH * NCODES)              // index_hist + softmax_hist contiguous
        out[OFF_IHIST + i] = 0.0f;
}

// -------- zero-fill the 268MB one-hot with 8B stores (base only 8B-aligned) ----------
__global__ __launch_bounds__(256) void vq_zero_onehot(float* __restrict__ out) {
    size_t i = (size_t)blockIdx.x * 256 + threadIdx.x;   // 33,554,432 float2's
    float2 z2; z2.x = 0.0f; z2.y = 0.0f;
    ((float2*)(out + OFF_ONEHOT))[i] = z2;
}

// -------- codebook squared norms ----------
__global__ __launch_bounds__(256) void vq_cnorm(const float* __restrict__ cb,
                                                float* __restrict__ cnorm) {
    int k = blockIdx.x * 256 + threadIdx.x;
    if (k < NCODES) {
        float s = 0.0f;
        for (int c = 0; c < EMBED; ++c) { float v = cb[(size_t)k * EMBED + c]; s += v * v; }
        cnorm[k] = s;
    }
}

// -------- main: WMMA distance GEMM + reductions + outputs ----------
__global__ __launch_bounds__(256) void vq_main(const float* __restrict__ z,
                                               const float* __restrict__ cb,
                                               const float* __restrict__ cnorm,
                                               float* __restrict__ out,
                                               float* __restrict__ acc) {
    extern __shared__ float smem[];
    float* zt      = smem;                                  // TILE_M * 130
    float* strip   = zt + TILE_M * STRIDE_ZT;               // TILE_M * 2052 (~263 KB)
    float* znorm   = strip + TILE_M * STRIDE_STRIP;         // 32
    float* row_min = znorm + TILE_M;                        // 32
    float* row_sum = row_min + TILE_M;                      // 32
    float* row_inv = row_sum + TILE_M;                      // 32
    int*   row_idx = (int*)(row_inv + TILE_M);              // 32
    float* red     = (float*)(row_idx + TILE_M);            // 256

    const int tid  = threadIdx.x;
    const int lane = tid & 31;
    const int wave = tid >> 5;        // 8 waves
    const int half = lane >> 4;       // 0: lanes 0-15, 1: lanes 16-31
    const int l16  = lane & 15;

    const int tile = blockIdx.x;          // 1024 tiles of 32 rows
    const int row0 = tile * TILE_M;
    const int b    = row0 >> 10;          // batch (tile never crosses batch)
    const int hw0  = row0 & 1023;

    // ---- stage z tile: zt[r][c] = z[b, c, hw0+r]; consecutive r = contiguous hw
    const float* zb = z + (size_t)b * (EMBED * HW) + hw0;
    for (int i = 0; i < 16; ++i) {
        int e = tid + i * 256;            // 4096 elems
        int r = e & 31, c = e >> 5;
        zt[r * STRIDE_ZT + c] = zb[(size_t)c * HW + r];
    }
    __syncthreads();
    if (tid < TILE_M) {
        float s = 0.0f;
        for (int c = 0; c < EMBED; ++c) { float v = zt[tid * STRIDE_ZT + c]; s += v * v; }
        znorm[tid] = s;
    }
    __syncthreads();

    // ---- pass 1: dist[32 x 2048] via chained V_WMMA_F32_16X16X4_F32
    // A frag (16x4 f32): lanes0-15 -> (K,K+1) of row l16 ; lanes16-31 -> (K+2,K+3)
    // B frag (4x16):     same pattern over codebook rows (row-major cb == B^T)
    // Each B fragment is reused for both 16-row M-subtiles (2 WMMAs / global load).
    for (int it = 0; it < 16; ++it) {
        const int nt   = wave + it * 8;           // code-tile 0..127
        const int code = nt * 16 + l16;
        const float* cbp = cb + (size_t)code * EMBED + half * 2;
        const float* ap0 = zt + l16 * STRIDE_ZT + half * 2;
        const float* ap1 = ap0 + SUB * STRIDE_ZT;
        v8f acc0 = {0.f, 0.f, 0.f, 0.f, 0.f, 0.f, 0.f, 0.f};
        v8f acc1 = {0.f, 0.f, 0.f, 0.f, 0.f, 0.f, 0.f, 0.f};
        #pragma unroll
        for (int kk = 0; kk < 32; ++kk) {
            v2f bv = *(const v2f*)(cbp + kk * 4);
            v2f a0 = *(const v2f*)(ap0 + kk * 4);
            v2f a1 = *(const v2f*)(ap1 + kk * 4);
            acc0 = __builtin_amdgcn_wmma_f32_16x16x4_f32(
                false, a0, false, bv, (short)0, acc0, false, false);
            acc1 = __builtin_amdgcn_wmma_f32_16x16x4_f32(
                false, a1, false, bv, (short)0, acc1, false, false);
        }
        const float cn = cnorm[code];
        #pragma unroll
        for (int j = 0; j < 8; ++j) {             // VGPR j: M = j + 8*half, N = l16
            int m = j + 8 * half;
            strip[m * STRIDE_STRIP + nt * 16 + l16] =
                znorm[m] + cn - 2.0f * acc0[j];
            strip[(m + SUB) * STRIDE_STRIP + nt * 16 + l16] =
                znorm[m + SUB] + cn - 2.0f * acc1[j];
        }
    }
    __syncthreads();

    // ---- pass 2: per-row min/argmin + sum of exp (wave32 reductions, 4 rows/wave)
    for (int rr = 0; rr < 4; ++rr) {
        const int r = wave * 4 + rr;
        const float* sr = strip + r * STRIDE_STRIP;
        float mv = 3.0e38f; int mi = 0;
        for (int k = lane; k < NCODES; k += 32) {
            float v = sr[k];
            if (v < mv) { mv = v; mi = k; }
        }
        #pragma unroll
        for (int off = 16; off > 0; off >>= 1) {
            float ov = __shfl_xor(mv, off, 32);
            int   oi = __shfl_xor(mi, off, 32);
            if (ov < mv || (ov == mv && oi < mi)) { mv = ov; mi = oi; }
        }
        float se = 0.0f;
        for (int k = lane; k < NCODES; k += 32) se += __expf(mv - sr[k]);
        #pragma unroll
        for (int off = 16; off > 0; off >>= 1) se += __shfl_xor(se, off, 32);
        if (lane == 0) { row_min[r] = mv; row_idx[r] = mi; row_sum[r] = se; }
    }
    __syncthreads();
    if (tid < TILE_M) row_inv[tid] = 1.0f / row_sum[tid];
    __syncthreads();

    // ---- pass 3: softmax histogram accumulation
    {
        float* shist = out + OFF_SHIST + (size_t)b * NCODES;
        for (int k = tid; k < NCODES; k += 256) {
            float s = 0.0f;
            #pragma unroll 4
            for (int r = 0; r < TILE_M; ++r)
                s += __expf(row_min[r] - strip[r * STRIDE_STRIP + k]) * row_inv[r];
            atomicAdd(&shist[k], s);
        }
    }

    // ---- pass 4: gather zq, write zq_ste, squared error, scatter outputs
    float lmse = 0.0f;
    for (int i = 0; i < 16; ++i) {
        int e = tid + i * 256;            // 4096 elems
        int r = e & 31, c = e >> 5;
        float q  = cb[(size_t)row_idx[r] * EMBED + c];
        float zv = zt[r * STRIDE_ZT + c];
        float d  = q - zv;
        lmse += d * d;
        out[OFF_ZQ + (size_t)b * (EMBED * HW) + (size_t)c * HW + hw0 + r] = q;
    }
    red[tid] = lmse;
    __syncthreads();
    #pragma unroll
    for (int s = 128; s > 0; s >>= 1) {
        if (tid < s) red[tid] += red[tid + s];
        __syncthreads();
    }
    if (tid == 0) atomicAdd(&acc[0], red[0]);

    if (tid < TILE_M) {
        int r = tid;
        int n = row0 + r;
        int idx = row_idx[r];
        out[OFF_IDX + n] = (float)idx;
        atomicAdd(&out[OFF_IHIST + (size_t)b * NCODES + idx], 1.0f);
        out[OFF_ONEHOT + (size_t)b * NCODES * HW + (size_t)idx * HW + hw0 + r] = 1.0f;
    }
    if (tid == 0) {
        float s = 0.0f;
        for (int r = 0; r < TILE_M; ++r) s += __logf(row_sum[r]);  // per-row softmax-loss term
        atomicAdd(&acc[1], s);
    }
}

// -------- finalize scalars + perplexity ----------
__global__ __launch_bounds__(256) void vq_finalize(const float* __restrict__ acc,
                                                   float* __restrict__ out) {
    __shared__ float red[256];
    int tid = threadIdx.x;
    float ent = 0.0f;
    for (int k = tid; k < NCODES; k += 256) {
        float cnt = 0.0f;
        for (int bb = 0; bb < BATCH; ++bb) cnt += out[OFF_IHIST + (size_t)bb * NCODES + k];
        float p = cnt * (1.0f / (float)NROWS);
        ent -= p * __logf(p + 1e-10f);
    }
    red[tid] = ent;
    __syncthreads();
    #pragma unroll
    for (int s = 128; s > 0; s >>= 1) {
        if (tid < s) red[tid] += red[tid + s];
        __syncthreads();
    }
    if (tid == 0) {
        float mse = acc[0] * (1.0f / 4194304.0f);   // mean over B*C*H*W
        float sl  = acc[1] * (1.0f / (float)NROWS);
        out[OFF_LOSS]   = mse + 0.25f * mse;        // quant + COMMIT_COST*commit (SOFTMAX_COST=0)
        out[OFF_PERP]   = __expf(red[0]);
        out[OFF_COMMIT] = mse;
        out[OFF_QUANT]  = mse;
        out[OFF_SLOSS]  = sl;
    }
}

extern "C" void kernel_launch(void* const* d_in, const int* in_sizes, int n_in,
                              void* d_out, int out_size, void* d_ws, size_t ws_size,
                              hipStream_t stream) {
    const float* z  = (const float*)d_in[0];   // [32,128,32,32]
    const float* cb = (const float*)d_in[1];   // [2048,128]
    float* out = (float*)d_out;
    float* cnorm = (float*)d_ws;               // 2048 floats
    float* acc   = cnorm + NCODES;             // [mse_sum, sloss_sum]

    vq_init<<<512, 256, 0, stream>>>(out, acc);
    vq_zero_onehot<<<131072, 256, 0, stream>>>(out);   // 33,554,432 float2 stores
    vq_cnorm<<<8, 256, 0, stream>>>(cb, cnorm);

    size_t smem = (size_t)(TILE_M * STRIDE_ZT + TILE_M * STRIDE_STRIP +
                           5 * TILE_M + 256) * sizeof(float);   // ~281 KB (CDNA5: 320KB/WGP)
    vq_main<<<NROWS / TILE_M, 256, smem, stream>>>(z, cb, cnorm, out, acc);

    vq_finalize<<<1, 256, 0, stream>>>(acc, out);
}